// EdgeGuidance_47313359733145
// MI455X (gfx1250) — compile-verified
//
#include <hip/hip_runtime.h>
#include <hip/hip_bf16.h>

// ---------------------------------------------------------------------------
// EdgeGuidance fused kernel for MI455X (gfx1250, wave32).
//
// One pass: RGB -> gray -> 5x5 Gaussian (reflect) -> Sobel (zero pad) ->
// magnitude -> 4x4 avg pool -> sigmoid^2.  Memory bound (~130 MB traffic at
// 23.3 TB/s ~ 5.6 us). Vertical blur runs on the matrix pipe via
// V_WMMA_F32_16X16X4_F32 banded matmuls (fp32, K=20 in 5 steps), co-executing
// with the VALU stencil work. Sobel zero-padding is applied by zeroing the
// (at most) one border row / column of the blurred LDS tile under
// block-uniform branches, so the hot pooling loop has no predication.
// ---------------------------------------------------------------------------

typedef float v2f __attribute__((ext_vector_type(2)));
typedef float v8f __attribute__((ext_vector_type(8)));

#define IMG_H 544
#define IMG_W 960
#define NBATCH 16
#define TILE_H 32           // mag rows per workgroup
#define TILE_W 64           // mag cols per workgroup

// gray region: rows [oy-3, oy+35), cols [ox-3, ox+67)
#define GR 38
#define GC 70
#define GSTR 72
// horizontal-blur region: rows [oy-3, oy+35) valid (38), cols [ox-1, ox+65)
// valid (66); padded up to 52x80 (3x16+4 rows, 5x16 cols) for WMMA tiling.
#define HR 52
#define HC 80
#define HSTR 84
// vertical-blur (fully blurred) region: 48x80 allocated, [0,34)x[0,66) used.
// Stride 84 floats = 336 B = 21*16 B -> every row is 16-byte aligned, so the
// pooling stage's 6-wide row reads can merge into ds_load_b128 + ds_load_b64.
#define VR 48
#define VSTR 84

// Gaussian(k=5, sigma=1.5), normalized.
#define G0 0.1200783f
#define G1 0.2338807f
#define G2 0.2920819f

__device__ __forceinline__ float band5(int d) {
    // A[m,k] of the banded vertical-blur matrix: weight G[d], d = k - m in [0,4]
    float v = 0.0f;
    v = (d == 0) ? G0 : v;
    v = (d == 1) ? G1 : v;
    v = (d == 2) ? G2 : v;
    v = (d == 3) ? G1 : v;
    v = (d == 4) ? G0 : v;
    return v;
}

__global__ __launch_bounds__(256)
void edge_guidance_fused(const float* __restrict__ in, float* __restrict__ out)
{
    __shared__ float s_gray[GR][GSTR];
    __shared__ float s_bh[HR][HSTR];
    __shared__ float s_bv[VR][VSTR];

    const int tid = threadIdx.x;
    const int ox  = blockIdx.x * TILE_W;
    const int oy  = blockIdx.y * TILE_H;
    const int b   = blockIdx.z;

    const float* __restrict__ base = in + (size_t)b * 3u * IMG_H * IMG_W;

    // ---------------- Stage 1: gray with reflect padding ----------------
    for (int idx = tid; idx < GR * GC; idx += 256) {
        int r = idx / GC;
        int c = idx - r * GC;
        int yy = oy - 3 + r;
        yy = (yy < 0) ? -yy : yy;
        yy = (yy >= IMG_H) ? (2 * IMG_H - 2 - yy) : yy;
        int xx = ox - 3 + c;
        xx = (xx < 0) ? -xx : xx;
        xx = (xx >= IMG_W) ? (2 * IMG_W - 2 - xx) : xx;
        size_t o = (size_t)yy * IMG_W + xx;
        float rv = base[o];
        float gv = base[(size_t)IMG_H * IMG_W + o];
        float bv = base[2u * (size_t)IMG_H * IMG_W + o];
        s_gray[r][c] = 0.2989f * rv + 0.587f * gv + 0.114f * bv;
    }
    __syncthreads();

    // ---------------- Stage 2: horizontal 5-tap blur ----------------
    // blurH row r  <-> image row oy-3+r   (valid r < 38, else 0)
    // blurH col c  <-> image col ox-1+c   (valid c < 66, else 0)
    for (int idx = tid; idx < HR * HC; idx += 256) {
        int r = idx / HC;
        int c = idx - r * HC;
        float v = 0.0f;
        if (r < GR && c < TILE_W + 2) {
            v = G0 * s_gray[r][c]     + G1 * s_gray[r][c + 1] +
                G2 * s_gray[r][c + 2] + G1 * s_gray[r][c + 3] +
                G0 * s_gray[r][c + 4];
        }
        s_bh[r][c] = v;
    }
    __syncthreads();

    // ---------------- Stage 3: vertical blur via WMMA banded matmul ------
    // D(16x16) = A(16x20 band) x B(20x16 blurH), K=20 in 5 steps of K=4.
    // A layout (32-bit 16x4): lanes 0-15 hold {K0,K1}, lanes 16-31 hold {K2,K3}.
    // B layout mirrors A's K placement; D: VGPR i = rows i / i+8.
    const int lane  = tid & 31;
    const int wave  = tid >> 5;
    const int m     = lane & 15;
    const int khalf = (lane < 16) ? 0 : 2;

    v2f afrag[5];
#pragma unroll
    for (int ks = 0; ks < 5; ++ks) {
        int kb = 4 * ks + khalf;
        afrag[ks].x = band5(kb - m);
        afrag[ks].y = band5(kb + 1 - m);
    }

    for (int t = wave; t < 15; t += 8) {   // 3x5 tiles of 16x16
        int ti = t / 5;
        int tj = t - ti * 5;
        int r0 = ti * 16;
        int c0 = tj * 16;
        v8f acc = {};
#pragma unroll
        for (int ks = 0; ks < 5; ++ks) {
            int krow = r0 + 4 * ks + khalf;
            v2f bfrag;
            bfrag.x = s_bh[krow][c0 + m];
            bfrag.y = s_bh[krow + 1][c0 + m];
            acc = __builtin_amdgcn_wmma_f32_16x16x4_f32(
                /*neg_a=*/false, afrag[ks],
                /*neg_b=*/false, bfrag,
                /*c_mod=*/(short)0, acc,
                /*reuse_a=*/false, /*reuse_b=*/false);
        }
        int rbase = r0 + ((lane < 16) ? 0 : 8);
#pragma unroll
        for (int iv = 0; iv < 8; ++iv)
            s_bv[rbase + iv][c0 + m] = acc[iv];
    }
    __syncthreads();

    // ---- Stage 3b: Sobel zero-padding fixup (block-uniform, border WGs) ----
    // s_bv row r <-> image row oy-1+r; col c <-> image col ox-1+c.
    // Rows/cols that fall outside the image must read as 0 for the Sobel
    // stage. Only row 0 (top tiles), row 33 (bottom tiles), col 0 (left
    // tiles), col 65 (right tiles) can be outside. Uniform scalar branches:
    // interior workgroups skip all four.
    if (oy == 0 && tid < 66)               s_bv[0][tid]  = 0.0f;
    if (oy + TILE_H == IMG_H && tid < 66)  s_bv[33][tid] = 0.0f;
    if (ox == 0 && tid < 34)               s_bv[tid][0]  = 0.0f;
    if (ox + TILE_W == IMG_W && tid < 34)  s_bv[tid][65] = 0.0f;
    __syncthreads();

    // ---------------- Stage 4: Sobel + magnitude + pool + sigmoid^2 ------
    if (tid < 128) {
        const int pr = tid >> 4;    // pooled row in tile [0,8)
        const int pc = tid & 15;    // pooled col in tile [0,16)

        float w[6][6];
#pragma unroll
        for (int r = 0; r < 6; ++r) {
#pragma unroll
            for (int c = 0; c < 6; ++c) {
                w[r][c] = s_bv[pr * 4 + r][pc * 4 + c];
            }
        }

        float sum = 0.0f;
#pragma unroll
        for (int iy = 0; iy < 4; ++iy) {
#pragma unroll
            for (int ix = 0; ix < 4; ++ix) {
                float gxv = (w[iy][ix + 2] + 2.0f * w[iy + 1][ix + 2] + w[iy + 2][ix + 2])
                          - (w[iy][ix]     + 2.0f * w[iy + 1][ix]     + w[iy + 2][ix]);
                float gyv = (w[iy + 2][ix] + 2.0f * w[iy + 2][ix + 1] + w[iy + 2][ix + 2])
                          - (w[iy][ix]     + 2.0f * w[iy][ix + 1]     + w[iy][ix + 2]);
                sum += sqrtf(gxv * gxv + gyv * gyv + 1e-6f);
            }
        }
        float down = sum * (1.0f / 16.0f);
        float z    = 5.0f * (down - 0.2f);
        float sig  = 1.0f / (1.0f + __expf(-z));

        const int PW = IMG_W / 4;  // 240
        const int PH = IMG_H / 4;  // 136
        size_t oidx = (size_t)b * PH * PW + (size_t)(oy / 4 + pr) * PW + (ox / 4 + pc);
        out[oidx] = sig * sig;
    }
}

extern "C" void kernel_launch(void* const* d_in, const int* in_sizes, int n_in,
                              void* d_out, int out_size, void* d_ws, size_t ws_size,
                              hipStream_t stream) {
    (void)in_sizes; (void)n_in; (void)out_size; (void)d_ws; (void)ws_size;
    const float* in = (const float*)d_in[0];
    float* out = (float*)d_out;
    dim3 grid(IMG_W / TILE_W, IMG_H / TILE_H, NBATCH);  // 15 x 17 x 16
    edge_guidance_fused<<<grid, 256, 0, stream>>>(in, out);
}